// Projector_30193620091588
// MI455X (gfx1250) — compile-verified
//
#include <hip/hip_runtime.h>
#include <math.h>

#define NUM_FACES 400000

typedef float v2f __attribute__((ext_vector_type(2)));
typedef float v8f __attribute__((ext_vector_type(8)));

// ---------------------------------------------------------------------------
// Kernel 1: zero the visibility table (1 byte per face, bit v = visible in view v)
// ---------------------------------------------------------------------------
__global__ void vis_zero_kernel(unsigned int* __restrict__ vis, int nwords) {
    int i = blockIdx.x * blockDim.x + threadIdx.x;
    if (i < nwords) vis[i] = 0u;
}

// ---------------------------------------------------------------------------
// Kernel 2: scatter pix_to_face into the packed per-view visibility table.
// pix_to_face is [V,S,S] int32 in [-1, NUM_FACES).
// ---------------------------------------------------------------------------
__global__ void vis_scatter_kernel(const int* __restrict__ p2f,
                                   unsigned int* __restrict__ vis,
                                   int SS, int total) {
    int i = blockIdx.x * blockDim.x + threadIdx.x;
    if (i >= total) return;
    int f = p2f[i];
    if (f < 0) return;
    int v = i / SS;                      // view index (0..3)
    atomicOr(&vis[f >> 2], 1u << ((f & 3) * 8 + v));
}

// ---------------------------------------------------------------------------
// Kernel 3: WMMA projection.  D = A x B with
//   A[16x4]  = homogeneous verts (x,y,z,1) for a tile of 16 vertices
//   B[4x16]  = columns 3v+j hold row j of (K*R | K*t) for view v  (cols 12..15 = 0)
//   D[16x16] = pix[m][3v+j]  (homogeneous pixel coords per view)
// Then per-vertex xy = (px/pz, py/pz) is written to workspace as [N][V] float2.
// ISA f32 layouts (CDNA5 ISA):
//   A: lane l, vgpr r -> M = l%16, K = 2*(l/16)+r
//   B: lane l, vgpr r -> N = l%16, K = 2*(l/16)+r
//   D: lane l, vgpr r -> M = r + 8*(l/16), N = l%16
// ---------------------------------------------------------------------------
__device__ __forceinline__ float proj_bval(const float* __restrict__ K,
                                           const float* __restrict__ R,
                                           const float* __restrict__ T,
                                           int col, int kk) {
    if (col >= 12) return 0.0f;
    int v = col / 3, j = col % 3;
    const float* Kr = K + v * 9 + j * 3;          // row j of K[v]
    if (kk < 3) {
        const float* Rv = R + v * 9;
        return Kr[0] * Rv[0 * 3 + kk] + Kr[1] * Rv[1 * 3 + kk] + Kr[2] * Rv[2 * 3 + kk];
    }
    const float* tv = T + v * 3;
    return Kr[0] * tv[0] + Kr[1] * tv[1] + Kr[2] * tv[2];
}

__global__ void project_wmma_kernel(const float* __restrict__ verts,
                                    const float* __restrict__ camR,
                                    const float* __restrict__ camT,
                                    const float* __restrict__ K,
                                    float* __restrict__ xyws,   // [N][4] float2
                                    int N) {
    __shared__ float tile_lds[8][16 * 17];        // 8 waves/block, stride-17 rows
    const int lane  = threadIdx.x & 31;
    const int wslot = threadIdx.x >> 5;
    const int wave  = (blockIdx.x * blockDim.x + threadIdx.x) >> 5;
    const int base  = wave * 16;
    if (base >= N) return;                        // wave-uniform exit: EXEC stays all-1s

    const int m  = lane & 15;                     // row (vertex within tile) / column
    const int hi = lane >> 4;

    // A operand: homogeneous vertex coords in the ISA 16x4 f32 layout.
    int vi = base + m; if (vi > N - 1) vi = N - 1;
    v2f a;
    if (hi == 0) { a.x = verts[vi * 3 + 0]; a.y = verts[vi * 3 + 1]; }
    else         { a.x = verts[vi * 3 + 2]; a.y = 1.0f; }

    // B operand: fused (K*R | K*t) per view, computed per-lane (uniform tiny reads).
    v2f b;
    b.x = proj_bval(K, camR, camT, m, hi * 2 + 0);
    b.y = proj_bval(K, camR, camT, m, hi * 2 + 1);

    v8f c = {0.0f, 0.0f, 0.0f, 0.0f, 0.0f, 0.0f, 0.0f, 0.0f};
    v8f d = __builtin_amdgcn_wmma_f32_16x16x4_f32(
        /*neg_a=*/false, a, /*neg_b=*/false, b,
        /*c_mod=*/(short)0, c, /*reuse_a=*/false, /*reuse_b=*/false);

    // Cross-lane redistribution of the D tile through LDS (intra-wave).
    float* tile = tile_lds[wslot];
#pragma unroll
    for (int r = 0; r < 8; ++r)
        tile[(r + 8 * hi) * 17 + m] = d[r];

    asm volatile("s_wait_dscnt 0" ::: "memory");  // wave-local LDS visibility

    if (hi == 0) {                                // lanes 0..15 emit vertex m
        int vert = base + m;
        if (vert < N) {
#pragma unroll
            for (int v = 0; v < 4; ++v) {
                float px = tile[m * 17 + 3 * v + 0];
                float py = tile[m * 17 + 3 * v + 1];
                float pz = tile[m * 17 + 3 * v + 2];
                float inv = 1.0f / pz;
                float2* o = (float2*)(xyws + (size_t)vert * 8 + v * 2);
                *o = make_float2(px * inv, py * inv);
            }
        }
    }
}

// ---------------------------------------------------------------------------
// Kernel 4: main gather / blend.  One wave per vertex, 2 channels per lane
// (C = 64).  Per-wave-uniform scalar state (face id, vis bits, xy) broadcasts
// in hardware.  Views with zero weight are skipped entirely (saves ~half of
// the random gathers).  Output [N][C] float, coalesced float2 stores.
// ---------------------------------------------------------------------------
__global__ void sample_blend_kernel(const int* __restrict__ faces,
                                    const float* __restrict__ feature,  // [V][C][S][S]
                                    const float* __restrict__ xyws,     // [N][V][2]
                                    const unsigned int* __restrict__ vis,
                                    float* __restrict__ out,            // [N][C]
                                    int N, int S, int C) {
    const int lane = threadIdx.x & 31;
    const int n    = (blockIdx.x * blockDim.x + threadIdx.x) >> 5;
    if (n >= N) return;

    const int SS = S * S;
    const int face = faces[n];
    unsigned vbits = (vis[face >> 2] >> ((face & 3) * 8)) & 0xFu;
    int wsum = __popc(vbits);
    unsigned active;
    float invw;
    if (wsum == 0) { active = 0xFu; invw = 0.25f; }       // all-invisible -> uniform weights
    else           { active = vbits; invw = 1.0f / (float)wsum; }

    const int c0 = lane * 2;
    float acc0 = 0.0f, acc1 = 0.0f;
    const float fS1 = (float)(S - 1);

#pragma unroll
    for (int v = 0; v < 4; ++v) {
        if (!((active >> v) & 1u)) continue;
        float x = xyws[(size_t)n * 8 + v * 2 + 0];
        float y = xyws[(size_t)n * 8 + v * 2 + 1];

        float x0f = floorf(x), y0f = floorf(y);
        float wx1 = x - x0f, wx0 = 1.0f - wx1;
        float wy1 = y - y0f, wy0 = 1.0f - wy1;

        float x1f = x0f + 1.0f, y1f = y0f + 1.0f;
        bool vx0 = (x0f >= 0.0f) & (x0f <= fS1);
        bool vx1 = (x1f >= 0.0f) & (x1f <= fS1);
        bool vy0 = (y0f >= 0.0f) & (y0f <= fS1);
        bool vy1 = (y1f >= 0.0f) & (y1f <= fS1);

        int xc0 = (int)fminf(fmaxf(x0f, 0.0f), fS1);
        int xc1 = (int)fminf(fmaxf(x1f, 0.0f), fS1);
        int yc0 = (int)fminf(fmaxf(y0f, 0.0f), fS1);
        int yc1 = (int)fminf(fmaxf(y1f, 0.0f), fS1);

        float w00 = wx0 * wy0 * (float)(vx0 & vy0);
        float w10 = wx1 * wy0 * (float)(vx1 & vy0);
        float w01 = wx0 * wy1 * (float)(vx0 & vy1);
        float w11 = wx1 * wy1 * (float)(vx1 & vy1);

        const float* fv0 = feature + ((size_t)v * C + c0) * SS;
        const float* fv1 = fv0 + SS;
        int i00 = yc0 * S + xc0, i10 = yc0 * S + xc1;
        int i01 = yc1 * S + xc0, i11 = yc1 * S + xc1;

        acc0 += w00 * fv0[i00] + w10 * fv0[i10] + w01 * fv0[i01] + w11 * fv0[i11];
        acc1 += w00 * fv1[i00] + w10 * fv1[i10] + w01 * fv1[i01] + w11 * fv1[i11];
    }

    float2* o = (float2*)(out + (size_t)n * C + c0);
    *o = make_float2(acc0 * invw, acc1 * invw);
}

// ---------------------------------------------------------------------------
// Launch
// ---------------------------------------------------------------------------
extern "C" void kernel_launch(void* const* d_in, const int* in_sizes, int n_in,
                              void* d_out, int out_size, void* d_ws, size_t ws_size,
                              hipStream_t stream) {
    const float* verts   = (const float*)d_in[0];
    const int*   faces   = (const int*)d_in[1];
    const float* cam_rot = (const float*)d_in[2];
    const float* cam_tr  = (const float*)d_in[3];
    const float* Kmat    = (const float*)d_in[4];
    const float* feature = (const float*)d_in[6];
    const int*   p2f     = (const int*)d_in[7];

    const int N  = in_sizes[1];            // 200000
    const int V  = in_sizes[3] / 3;        // 4
    const int SS = in_sizes[7] / V;        // S*S
    const int S  = (int)(sqrt((double)SS) + 0.5);
    const int C  = (int)((long long)in_sizes[6] / ((long long)V * SS));  // 64

    // Workspace layout: [vis table words | xy (N x V x float2)]
    const int visWords = (NUM_FACES + 3) / 4;
    unsigned int* vis = (unsigned int*)d_ws;
    float* xyws = (float*)((char*)d_ws + (((size_t)visWords * 4 + 255) & ~(size_t)255));

    // 1) zero visibility table
    {
        int threads = 256, blocks = (visWords + threads - 1) / threads;
        vis_zero_kernel<<<blocks, threads, 0, stream>>>(vis, visWords);
    }
    // 2) scatter pix_to_face bits
    {
        int total = V * SS;
        int threads = 256, blocks = (total + threads - 1) / threads;
        vis_scatter_kernel<<<blocks, threads, 0, stream>>>(p2f, vis, SS, total);
    }
    // 3) WMMA projection (16 vertices per wave, 8 waves per block)
    {
        int tiles = (N + 15) / 16;         // one wave per tile
        int threads = 256;
        int blocks = (tiles * 32 + threads - 1) / threads;
        project_wmma_kernel<<<blocks, threads, 0, stream>>>(verts, cam_rot, cam_tr, Kmat,
                                                            xyws, N);
    }
    // 4) gather + blend (one wave per vertex)
    {
        long long totalThreads = (long long)N * 32;
        int threads = 256;
        int blocks = (int)((totalThreads + threads - 1) / threads);
        sample_blend_kernel<<<blocks, threads, 0, stream>>>(faces, feature, xyws, vis,
                                                            (float*)d_out, N, S, C);
    }
}